// NCJT_RxUE_8950711845842
// MI455X (gfx1250) — compile-verified
//
#include <hip/hip_runtime.h>
#include <math.h>

// CDNA5 / gfx1250, wave32.
typedef __attribute__((ext_vector_type(2))) float v2f;
typedef __attribute__((ext_vector_type(8))) float v8f;

// One thread = one position p = ((bf*6 + pair)*2 + stream).
// P is an exact multiple of 256 -> no tail, EXEC always all-ones at the WMMAs.
__global__ __launch_bounds__(256) void ncjt_rxue_kernel(
    const float* __restrict__ ryR, const float* __restrict__ ryI,
    const float* __restrict__ hR,  const float* __restrict__ hI,
    float* __restrict__ bits_out, float* __restrict__ gains_out,
    float* __restrict__ nvar_out)
{
    const int lane = threadIdx.x & 31;
    const int gpos = blockIdx.x * 256 + threadIdx.x;     // position id
    const int strm = gpos & 1;
    const int pr   = gpos >> 1;                          // bf*6 + pair
    const int pair = pr % 6;
    const int bf   = pr / 6;
    // DATA_SYMS pairs: d0 in {0,3,5,7,9,12}, d1 = d0+1  (no table -> no divergent loads)
    const int d0 = 2 * pair + (pair > 0 ? 1 : 0) + (pair == 5 ? 1 : 0);
    const int d1 = d0 + 1;

    // ---- per-lane constellation constants (label c = lane&15, Gray 16-QAM) ----
    const int   c  = lane & 15;
    const int   b0 = (c >> 3) & 1, b1 = (c >> 2) & 1, b2 = (c >> 1) & 1, b3 = c & 1;
    const float Ip = (float)((1 - 2 * b0) * (1 + 2 * b2));   // PAM level in {-3,-1,1,3}
    const float Qp = (float)((1 - 2 * b1) * (1 + 2 * b3));
    const float inv_s = 0.3162277660168379f;                 // 1/sqrt(10)
    const float Cr = Ip * inv_s;
    const float Ci = Qp * inv_s;
    const float Cm = (Ip * Ip + Qp * Qp) * 0.1f;             // |C_c|^2
    // A matrix = constellation, 16x4 (M=label c, K=4), all compile-time per lane:
    //   A[c,:] = (Cr_c, Ci_c, 1, |C_c|^2)
    //   layout: VGPR0 = K0 (lanes 0-15, M=lane) / K2 (lanes 16-31, M=lane-16)
    //           VGPR1 = K1 / K3
    v2f Am;
    Am.x = (lane < 16) ? Cr : 1.0f;
    Am.y = (lane < 16) ? Ci : Cm;

    // ---- streaming combine: B128 loads ----
    const float4* ryR4 = (const float4*)ryR;
    const float4* ryI4 = (const float4*)ryI;
    const float4* hR4  = (const float4*)hR;
    const float4* hI4  = (const float4*)hI;

    const int ryb0 = bf * 14 + d0;          // float4 index into (..,14,4,1)
    const int ryb1 = bf * 14 + d1;
    const int hb0  = ryb0 * 4;              // float4 index into (..,14,4,4)
    const int hb1  = ryb1 * 4;

    const float4 y1Rv = ryR4[ryb0];
    const float4 y1Iv = ryI4[ryb0];
    const float4 y2Rv = ryR4[ryb1];
    const float4 y2Iv = ryI4[ryb1];
    const float y1r[4] = {y1Rv.x, y1Rv.y, y1Rv.z, y1Rv.w};
    const float y1i[4] = {y1Iv.x, y1Iv.y, y1Iv.z, y1Iv.w};
    const float y2r[4] = {y2Rv.x, y2Rv.y, y2Rv.z, y2Rv.w};
    const float y2i[4] = {y2Iv.x, y2Iv.y, y2Iv.z, y2Iv.w};

    float xRe = 0.f, xIm = 0.f, g = 0.f;
#pragma unroll
    for (int r = 0; r < 4; ++r) {
        const float4 hr0 = hR4[hb0 + r];    // cols 0..3, sym d0, antenna r (real)
        const float4 hr1 = hR4[hb1 + r];
        const float4 hi0 = hI4[hb0 + r];
        const float4 hi1 = hI4[hb1 + r];
        // h_avg = 0.5*(h_d0 + h_d1); h1 = col0+col2, h2 = col1+col3
        const float h1R = 0.5f * (hr0.x + hr1.x + hr0.z + hr1.z);
        const float h2R = 0.5f * (hr0.y + hr1.y + hr0.w + hr1.w);
        const float h1I = 0.5f * (hi0.x + hi1.x + hi0.z + hi1.z);
        const float h2I = 0.5f * (hi0.y + hi1.y + hi0.w + hi1.w);
        // stream0: x1 = conj(h1)*y1 + h2*conj(y2)   (a=h1, b=h2)
        // stream1: x2 = conj(h2)*y1 - h1*conj(y2)   (a=h2, b=-h1)
        const float aR = strm ? h2R : h1R;
        const float aI = strm ? h2I : h1I;
        const float bR = strm ? -h1R : h2R;
        const float bI = strm ? -h1I : h2I;
        xRe += aR * y1r[r] + aI * y1i[r] + bR * y2r[r] + bI * y2i[r];
        xIm += aR * y1i[r] - aI * y1r[r] + bI * y2r[r] - bR * y2i[r];
        g   += h1R * h1R + h1I * h1I + h2R * h2R + h2I * h2I;
    }

    const float yR  = xRe / g;
    const float yI  = xIm / g;
    const float mag = yR * yR + yI * yI;
    const float nR  = -2.0f * yR;
    const float nI  = -2.0f * yI;

    if (strm == 0) gains_out[pr] = g;       // gains shape (B,F,6)

    // ---- distance matrix via V_WMMA_F32_16X16X4_F32 (transposed mapping) ----
    // D[c,p] = Cr_c*(-2yR_p) + Ci_c*(-2yI_p) + 1*|y_p|^2 + |C_c|^2*1 = |y_p - C_c|^2
    // B matrix = positions, 4x16 (K=4, N=position):
    //   VGPR0 = K0 (lanes 0-15: -2yR, col=lane) / K2 (lanes 16-31: |y|^2, col=lane-16)
    //   VGPR1 = K1 (-2yI) / K3 (1.0)
    // D layout: lane l holds col p=l&15, rows c = v + (l<16 ? 0 : 8) in 8 VGPRs ->
    // per-position argmin is 7 in-register mins + ONE cross-half shuffle.
    unsigned tileKey[2];
#pragma unroll
    for (int t = 0; t < 2; ++t) {
        const int src = t * 16 + (lane & 15);     // wave-local position for this column
        const float sR = __shfl(nR, src);
        const float sI = __shfl(nI, src);
        const float sM = __shfl(mag, src);
        v2f Bm;
        Bm.x = (lane < 16) ? sR : sM;             // K0 : K2
        Bm.y = (lane < 16) ? sI : 1.0f;           // K1 : K3
        v8f Cz = {};
        const v8f d = __builtin_amdgcn_wmma_f32_16x16x4_f32(
            /*neg_a=*/false, Am, /*neg_b=*/false, Bm,
            /*c_mod=*/(short)0, Cz, /*reuse_a=*/false, /*reuse_b=*/false);

        // local argmin over this lane-half's 8 labels; label packed in low 4 bits.
        // fmaxf(d,0): WMMA rounding can give tiny negative d whose IEEE bits would
        // order as a huge unsigned -> clamp keeps unsigned compare == float compare.
        const int labBase = (lane < 16) ? 0 : 8;
        unsigned key = 0xFFFFFFFFu;
#pragma unroll
        for (int v = 0; v < 8; ++v) {
            const unsigned k = (__float_as_uint(fmaxf(d[v], 0.0f)) & 0xFFFFFFF0u)
                             | (unsigned)(labBase + v);
            key = (k < key) ? k : key;
        }
        // merge the two halves (labels 0-7 vs 8-15) for the same column p
        const unsigned ok = __shfl_xor(key, 16);
        tileKey[t] = (ok < key) ? ok : key;
    }

    // Tile 0 columns are owned by lanes 0-15, tile 1 columns by lanes 16-31,
    // and after the half-merge every lane holds its own column's argmin.
    const unsigned myKey = (lane < 16) ? tileKey[0] : tileKey[1];
    const int myLab = (int)(myKey & 15u);

    // bits: BITMAT[i][lab] = (lab >> (3-i)) & 1, written as float4 (coalesced)
    float4 bv;
    bv.x = (float)((myLab >> 3) & 1);
    bv.y = (float)((myLab >> 2) & 1);
    bv.z = (float)((myLab >> 1) & 1);
    bv.w = (float)(myLab & 1);
    ((float4*)bits_out)[gpos] = bv;

    if (gpos == 0) *nvar_out = 0.05f;
}

extern "C" void kernel_launch(void* const* d_in, const int* in_sizes, int n_in,
                              void* d_out, int out_size, void* d_ws, size_t ws_size,
                              hipStream_t stream) {
    (void)n_in; (void)out_size; (void)d_ws; (void)ws_size;
    const float* ryR = (const float*)d_in[0];
    const float* ryI = (const float*)d_in[1];
    const float* hR  = (const float*)d_in[2];
    const float* hI  = (const float*)d_in[3];

    const int  BF     = in_sizes[0] / 56;        // 14 syms * 4 antennas
    const long bitsN  = (long)BF * 48;           // (B,F,6,8)
    const long gainsN = (long)BF * 6;            // (B,F,6)

    float* bits  = (float*)d_out;
    float* gains = bits + bitsN;
    float* nvar  = bits + bitsN + gainsN;

    const int P      = BF * 12;                  // positions: 6 pairs * 2 streams
    const int blocks = P / 256;                  // exact: P = 1,572,864 -> 6144

    ncjt_rxue_kernel<<<blocks, 256, 0, stream>>>(ryR, ryI, hR, hI, bits, gains, nvar);
}